// EMLTree1D_19636590477917
// MI455X (gfx1250) — compile-verified
//
#include <hip/hip_runtime.h>
#include <math.h>

typedef __attribute__((ext_vector_type(2))) float v2f;
typedef __attribute__((ext_vector_type(8))) float v8f;

#define BLOCK      128
#define NLEAF      64
#define NPAIR      63
#define LVL_STRIDE 65   // pad 64 -> 65 dwords: conflict-free per-thread column reads

static __device__ __forceinline__ double clampd(double v) {
  const double C = 1e300;
  if (v != v) return 0.0;        // NaN -> 0
  if (v >  C) return  C;         // +inf / big -> +C
  if (v < -C) return -C;         // -inf / small -> -C
  return v;
}

// Reference _blend with x_i == 0 (x is real):
//   blend_r = p_const + p_child_safe*child_r + x0*p1 + x1*p2
//   blend_i = p_child_safe*child_i
//   then sequential bypass overrides: const -> 1, var v -> x_v
static __device__ __forceinline__ void blendf(const double* __restrict__ ps,
                                              double x0, double x1,
                                              double cr, double ci,
                                              double& br, double& bi) {
  const double CHILD_EPS = 2.220446049250313e-16;   // DBL_EPSILON
  const double BYPASS    = 1.0 - 2.220446049250313e-16;
  double pc  = ps[3];
  bool   mc  = pc > CHILD_EPS;
  double pcs = mc ? pc : 0.0;
  double scr = mc ? cr : 0.0;
  double sci = mc ? ci : 0.0;
  br = ps[0] + pcs * scr + x0 * ps[1] + x1 * ps[2];
  bi = pcs * sci;
  if (ps[0] > BYPASS) { br = 1.0; bi = 0.0; }
  if (ps[1] > BYPASS) { br = x0;  bi = 0.0; }
  if (ps[2] > BYPASS) { br = x1;  bi = 0.0; }
}

__global__ void __launch_bounds__(BLOCK)
emltree_kernel(const float* __restrict__ x,
               const double* __restrict__ leaf_logits,
               const double* __restrict__ gate_logits,
               double* __restrict__ out, int n) {
  __shared__ float  s_level[BLOCK * LVL_STRIDE];  // 33280 B
  __shared__ double s_gate[NPAIR * 8];            //  4032 B  [pair][side][4]
  __shared__ float  s_wf[NLEAF * 3];              //   768 B  leaf probs (f32 for WMMA B)

  const int tid  = threadIdx.x;
  const int lane = tid & 31;
  const int wave = tid >> 5;
  const long long base = (long long)blockIdx.x * BLOCK;

  // Warm L2/WGP$ for this block's slice of x (gfx1250 global_prefetch_b8).
  if (base + tid < n) __builtin_prefetch(&x[(base + tid) * 2], 0, 3);

  // ---------- Phase 0: softmaxes (tau = 1) ----------
  if (tid < NLEAF) {
    const double* L = leaf_logits + tid * 3;
    double m  = fmax(L[0], fmax(L[1], L[2]));
    double e0 = exp(L[0] - m), e1 = exp(L[1] - m), e2 = exp(L[2] - m);
    double s  = e0 + e1 + e2;
    double p0 = e0 / s, p1 = e1 / s, p2 = e2 / s;
    s_wf[tid*3+0] = (float)p0;
    s_wf[tid*3+1] = (float)p1;
    s_wf[tid*3+2] = (float)p2;
    if (blockIdx.x == 0) {                 // leaf_probs output (64x3 f64)
      double* o = out + 2LL * n + (long long)tid * 3;
      o[0] = p0; o[1] = p1; o[2] = p2;
    }
  }
  if (tid < NPAIR * 2) {                   // 126 rows of 4
    const double* G = gate_logits + tid * 4;
    double m  = fmax(fmax(G[0], G[1]), fmax(G[2], G[3]));
    double e0 = exp(G[0]-m), e1 = exp(G[1]-m), e2 = exp(G[2]-m), e3 = exp(G[3]-m);
    double s  = e0 + e1 + e2 + e3;
    double p0 = e0/s, p1 = e1/s, p2 = e2/s, p3 = e3/s;
    s_gate[tid*4+0] = p0; s_gate[tid*4+1] = p1;
    s_gate[tid*4+2] = p2; s_gate[tid*4+3] = p3;
    if (blockIdx.x == 0) {                 // gate_probs output (63x2x4 f64)
      double* o = out + 2LL * n + NLEAF * 3 + (long long)tid * 4;
      o[0] = p0; o[1] = p1; o[2] = p2; o[3] = p3;
    }
  }
  __syncthreads();

  // ---------- Phase 1: level0 = [1,x0,x1,0] @ wT via V_WMMA_F32_16X16X4_F32 ----------
  // Each wave covers 32 batch rows (2 passes of 16); 4 WMMAs cover the 64 leaves.
  for (int pass = 0; pass < 2; ++pass) {
    const int mloc0 = wave * 32 + pass * 16;                // local row base of tile
    const int mrow  = (lane < 16) ? lane : (lane - 16);
    const long long grow = base + mloc0 + mrow;
    // A (16x4, MxK): lanes 0-15 hold {K0,K1} = {1, x0}; lanes 16-31 hold {K2,K3} = {x1, 0}
    v2f a;
    if (grow < n) {
      if (lane < 16) a = (v2f){1.0f, x[grow*2 + 0]};
      else           a = (v2f){x[grow*2 + 1], 0.0f};
    } else {
      a = (v2f){0.0f, 0.0f};
    }
#pragma unroll
    for (int g = 0; g < 4; ++g) {
      const int nn = g * 16 + (lane & 15);                  // leaf (column) index
      // B (4x16, KxN): lanes 0-15 hold {K0,K1}; lanes 16-31 hold {K2,K3(=0 pad)}
      v2f b;
      if (lane < 16) b = (v2f){ s_wf[nn*3+0], s_wf[nn*3+1] };
      else           b = (v2f){ s_wf[nn*3+2], 0.0f };
      v8f c = {};
      v8f d = __builtin_amdgcn_wmma_f32_16x16x4_f32(
                  /*neg_a=*/false, a, /*neg_b=*/false, b,
                  /*c_mod=*/(short)0, c, /*reuse_a=*/false, /*reuse_b=*/false);
      // D (16x16 f32): VGPR r, lanes 0-15 -> M=r, lanes 16-31 -> M=r+8
#pragma unroll
      for (int r = 0; r < 8; ++r) {
        const int m = mloc0 + r + ((lane < 16) ? 0 : 8);
        s_level[m * LVL_STRIDE + nn] = d[r];
      }
    }
  }
  __syncthreads();

  // ---------- Phase 2: fp64 complex tree combine, depth-first (7-deep stack) ----------
  const long long row = base + tid;
  if (row < n) {
    const double x0 = (double)x[row*2 + 0];
    const double x1 = (double)x[row*2 + 1];
    double sr[7], si[7];
    int sp = 0;
    for (int t = 0; t < NLEAF; ++t) {
      sr[sp] = (double)s_level[tid * LVL_STRIDE + t];
      si[sp] = 0.0;
      ++sp;
      int tt = t, k = 0;
      while (tt & 1) {
        // merging level-k pair j = t>>(k+1); gate offset for level k = 64 - (64>>k)
        const int gi = (64 - (64 >> k)) + (t >> (k + 1));
        const double* pg = &s_gate[gi * 8];
        double Lr, Li, Rr, Ri;
        blendf(pg,     x0, x1, sr[sp-2], si[sp-2], Lr, Li);   // left,  side 0
        blendf(pg + 4, x0, x1, sr[sp-1], si[sp-1], Rr, Ri);   // right, side 1
        // exp(L) - log(R), complex
        const double er  = exp(Lr);
        const double cre = er * cos(Li);
        const double cim = er * sin(Li);
        const double lgr = 0.5 * log(Rr*Rr + Ri*Ri);
        const double lgi = atan2(Ri, Rr);
        --sp;
        sr[sp-1] = clampd(cre - lgr);
        si[sp-1] = clampd(cim - lgi);
        ++k; tt >>= 1;
      }
    }
    out[row*2 + 0] = sr[0];   // complex128 interleaved (re, im)
    out[row*2 + 1] = si[0];
  }
}

extern "C" void kernel_launch(void* const* d_in, const int* in_sizes, int n_in,
                              void* d_out, int out_size, void* d_ws, size_t ws_size,
                              hipStream_t stream) {
  const float*  x    = (const float*) d_in[0];   // (B, 2) f32
  const double* leaf = (const double*)d_in[1];   // (64, 3) f64
  const double* gate = (const double*)d_in[2];   // (63, 2, 4) f64
  double* out = (double*)d_out;                  // [B complex128] + [192 f64] + [504 f64]
  const int n = in_sizes[0] / 2;
  const int grid = (n + BLOCK - 1) / BLOCK;
  hipLaunchKernelGGL(emltree_kernel, dim3(grid), dim3(BLOCK), 0, stream,
                     x, leaf, gate, out, n);
}